// PointNetSaModule_23304492548690
// MI455X (gfx1250) — compile-verified
//
#include <hip/hip_runtime.h>
#include <hip/hip_bf16.h>

typedef __attribute__((ext_vector_type(2))) float v2f;
typedef __attribute__((ext_vector_type(8))) float v8f;

constexpr int BB   = 4;
constexpr int NN   = 16384;
constexpr int SS   = 4096;
constexpr int KK   = 32;
constexpr int CINC = 16;
constexpr float EPSV = 1e-5f;

// ---------------------------------------------------------------------------
// Kernel 1: per-point squared norms
// ---------------------------------------------------------------------------
__global__ void __launch_bounds__(256) norms_kernel(const float* __restrict__ xyz,
                                                    float* __restrict__ pnorm) {
    int i = blockIdx.x * 256 + threadIdx.x;   // over BB*NN
    if (i < BB * NN) {
        float x = xyz[3 * i + 0], y = xyz[3 * i + 1], z = xyz[3 * i + 2];
        pnorm[i] = x * x + y * y + z * z;
    }
}

// ---------------------------------------------------------------------------
// Kernel 2: gather sampled centers; write new_xyz + sample_idx outputs
// ---------------------------------------------------------------------------
__global__ void __launch_bounds__(256) gather_kernel(const float* __restrict__ xyz,
                                                     const int* __restrict__ sidx,
                                                     float* __restrict__ qxyz,
                                                     float* __restrict__ qnorm,
                                                     float* __restrict__ out_xyz,
                                                     float* __restrict__ out_idx) {
    int i = blockIdx.x * 256 + threadIdx.x;   // over BB*SS
    if (i < BB * SS) {
        int b = i / SS;
        int n = sidx[i];
        const float* p = xyz + (size_t)(b * NN + n) * 3;
        float x = p[0], y = p[1], z = p[2];
        qxyz[3 * i + 0] = x; qxyz[3 * i + 1] = y; qxyz[3 * i + 2] = z;
        out_xyz[3 * i + 0] = x; out_xyz[3 * i + 1] = y; out_xyz[3 * i + 2] = z;
        qnorm[i] = x * x + y * y + z * z;
        out_idx[i] = (float)n;
    }
}

// ---------------------------------------------------------------------------
// Kernel 3: top-K (K smallest distances) per query.
// One 256-thread block per query; each thread owns 64 points (dist in regs,
// 64-bit exclusion mask), 32 rounds of register scan + LDS tree argmin.
// ---------------------------------------------------------------------------
__global__ void __launch_bounds__(256) topk_kernel(const float* __restrict__ xyz,
                                                   const float* __restrict__ pnorm,
                                                   const float* __restrict__ qxyz,
                                                   const float* __restrict__ qnorm,
                                                   int* __restrict__ topk) {
    __shared__ float sval[256];
    __shared__ int   spay[256];

    int q = blockIdx.x;          // over BB*SS
    int b = q / SS;
    int tid = threadIdx.x;

    float qx = qxyz[3 * q + 0], qy = qxyz[3 * q + 1], qz = qxyz[3 * q + 2];
    float qn = qnorm[q];
    const float* px = xyz + (size_t)b * NN * 3;
    const float* pn = pnorm + (size_t)b * NN;

    __builtin_prefetch(px + 3 * tid, 0, 0);       // global_prefetch_b8
    __builtin_prefetch(pn + tid, 0, 0);

    float dist[64];
#pragma unroll
    for (int j = 0; j < 64; ++j) {
        int n = tid + 256 * j;
        float x = px[3 * n + 0], y = px[3 * n + 1], z = px[3 * n + 2];
        dist[j] = qn + pn[n] - 2.0f * (qx * x + qy * y + qz * z);
    }

    unsigned m0 = 0u, m1 = 0u;   // exclusion mask for this thread's 64 points

    for (int r = 0; r < KK; ++r) {
        float best = 3.4e38f;
        int bj = 0;
#pragma unroll
        for (int j = 0; j < 64; ++j) {
            bool used = (j < 32) ? ((m0 >> j) & 1u) : ((m1 >> (j - 32)) & 1u);
            bool take = (!used) && (dist[j] < best);
            best = take ? dist[j] : best;
            bj   = take ? j : bj;
        }
        sval[tid] = best;
        spay[tid] = (bj << 8) | tid;
        __syncthreads();
        for (int off = 128; off > 0; off >>= 1) {
            if (tid < off) {
                if (sval[tid + off] < sval[tid]) {
                    sval[tid] = sval[tid + off];
                    spay[tid] = spay[tid + off];
                }
            }
            __syncthreads();
        }
        int pay  = spay[0];
        int wtid = pay & 255;
        int wj   = pay >> 8;
        if (tid == wtid) {
            if (wj < 32) m0 |= (1u << wj); else m1 |= (1u << (wj - 32));
        }
        if (tid == 0) topk[(size_t)q * KK + r] = wtid + 256 * wj;
        __syncthreads();
    }
}

// ---------------------------------------------------------------------------
// Fused linear + BN + ReLU layer via V_WMMA_F32_16X16X4_F32.
//   Xs : 32 x (4*NK) activations in LDS (row stride xstride, even)
//   Wt : transposed zero-padded weights in LDS, Wt[ch*wstride + k]
//   Ys : 32 x (16*NT) output activations in LDS
// A layout (16x4 f32): lanes 0-15 -> M=lane,   VGPR0/1 = K 0/1;
//                      lanes 16-31 -> M=lane-16, VGPR0/1 = K 2/3.
// B layout (4x16):     lanes 0-15 -> N=lane, rows K 0/1; lanes 16-31 rows 2/3.
// D layout (16x16):    VGPR i: lanes 0-15 -> (M=i, N=lane),
//                              lanes 16-31 -> (M=i+8, N=lane-16).
// ---------------------------------------------------------------------------
template <int NK, int NT>
__device__ inline void mlp_layer(const float* Xs, int xstride,
                                 const float* Wt, int wstride,
                                 const float* __restrict__ bias,
                                 const float* __restrict__ g,
                                 const float* __restrict__ bt,
                                 const float* __restrict__ rm,
                                 const float* __restrict__ rv,
                                 float* Ys, int ystride, int lane) {
    const int l16 = lane & 15;
    const int hi  = lane >> 4;       // 0 for lanes 0-15, 1 for lanes 16-31

    // Folded BN parameters; channel owned by this lane in column tile t.
    float sc[NT], sh[NT], bv[NT];
#pragma unroll
    for (int t = 0; t < NT; ++t) {
        const int ch = t * 16 + l16;
        const float inv = rsqrtf(rv[ch] + EPSV);
        sc[t] = g[ch] * inv;
        sh[t] = bt[ch] - rm[ch] * sc[t];
        bv[t] = bias[ch];
    }

    for (int rt = 0; rt < 2; ++rt) {           // two 16-row tiles (32 neighbors)
        const int row = rt * 16 + l16;
        v2f a[NK];                             // A fragments, invariant over t
#pragma unroll
        for (int kk = 0; kk < NK; ++kk)
            a[kk] = *(const v2f*)&Xs[row * xstride + kk * 4 + hi * 2];

#pragma unroll
        for (int t = 0; t < NT; ++t) {         // 16-column tiles
            const int ch = t * 16 + l16;
            v8f acc = {bv[t], bv[t], bv[t], bv[t], bv[t], bv[t], bv[t], bv[t]};
#pragma unroll
            for (int kk = 0; kk < NK; ++kk) {
                v2f wb = *(const v2f*)&Wt[ch * wstride + kk * 4 + hi * 2];
                acc = __builtin_amdgcn_wmma_f32_16x16x4_f32(
                        false, a[kk], false, wb, (short)0, acc, false, false);
            }
#pragma unroll
            for (int i = 0; i < 8; ++i) {
                float v = acc[i] * sc[t] + sh[t];   // folded batch-norm
                v = fmaxf(v, 0.0f);                 // ReLU
                Ys[(rt * 16 + hi * 8 + i) * ystride + ch] = v;
            }
        }
    }
}

// ---------------------------------------------------------------------------
// Kernel 4: per-query fused MLP (19->32->32->64) + max over K neighbors.
// One wave (32 lanes) per query; lane k gathers neighbor k.
// Weights staged once into LDS (transposed + zero-padded).
// ---------------------------------------------------------------------------
__global__ void __launch_bounds__(32) mlp_kernel(
        const float* __restrict__ xyz, const float* __restrict__ feat,
        const int* __restrict__ topk, const float* __restrict__ qxyz,
        const float* __restrict__ w0, const float* __restrict__ b0,
        const float* __restrict__ g0, const float* __restrict__ bt0,
        const float* __restrict__ rm0, const float* __restrict__ rv0,
        const float* __restrict__ w1, const float* __restrict__ b1,
        const float* __restrict__ g1, const float* __restrict__ bt1,
        const float* __restrict__ rm1, const float* __restrict__ rv1,
        const float* __restrict__ w2, const float* __restrict__ b2,
        const float* __restrict__ g2, const float* __restrict__ bt2,
        const float* __restrict__ rm2, const float* __restrict__ rv2,
        float* __restrict__ out_feat) {
    __shared__ float Xs[32 * 20];    // layer-0 input: 19 channels + zero pad
    __shared__ float Ys[32 * 34];    // layer-0 output (even stride -> b64 loads)
    __shared__ float Zs[32 * 34];    // layer-1 output
    __shared__ float Fs[32 * 64];    // layer-2 output
    __shared__ float Wt0[32 * 20];   // w0^T, zero-padded k=19
    __shared__ float Wt1[32 * 32];   // w1^T
    __shared__ float Wt2[64 * 32];   // w2^T

    const int q    = blockIdx.x;     // over BB*SS
    const int b    = q / SS;
    const int lane = threadIdx.x;

    // Stage transposed, padded weights into LDS (cooperative).
    for (int idx = lane; idx < 32 * 20; idx += 32) {
        const int ch = idx / 20, k = idx % 20;
        Wt0[idx] = (k < 19) ? w0[k * 32 + ch] : 0.0f;
    }
    for (int idx = lane; idx < 32 * 32; idx += 32) {
        const int ch = idx / 32, k = idx % 32;
        Wt1[idx] = w1[k * 32 + ch];
    }
    for (int idx = lane; idx < 64 * 32; idx += 32) {
        const int ch = idx / 32, k = idx % 32;
        Wt2[idx] = w2[k * 64 + ch];
    }

    // Gather: lane k builds row k = [xyz_diff(3) | feature(16) | pad]
    const int n = topk[(size_t)q * KK + lane];
    const float qx = qxyz[3 * q + 0], qy = qxyz[3 * q + 1], qz = qxyz[3 * q + 2];
    const float* p = xyz + (size_t)(b * NN + n) * 3;
    Xs[lane * 20 + 0] = p[0] - qx;
    Xs[lane * 20 + 1] = p[1] - qy;
    Xs[lane * 20 + 2] = p[2] - qz;
    const float* f = feat + (size_t)(b * NN + n) * CINC;
#pragma unroll
    for (int c = 0; c < CINC; ++c) Xs[lane * 20 + 3 + c] = f[c];
    Xs[lane * 20 + 19] = 0.0f;
    __syncthreads();

    mlp_layer<5, 2>(Xs, 20, Wt0, 20, b0, g0, bt0, rm0, rv0, Ys, 34, lane);
    __syncthreads();
    mlp_layer<8, 2>(Ys, 34, Wt1, 32, b1, g1, bt1, rm1, rv1, Zs, 34, lane);
    __syncthreads();
    mlp_layer<8, 4>(Zs, 34, Wt2, 32, b2, g2, bt2, rm2, rv2, Fs, 64, lane);
    __syncthreads();

    // Max-pool over the 32 neighbors; lane handles channels lane, lane+32.
#pragma unroll
    for (int cc = 0; cc < 2; ++cc) {
        const int ch = lane + cc * 32;
        float m = 0.0f;                      // post-ReLU activations are >= 0
        for (int r = 0; r < 32; ++r) {
            float v = Fs[r * 64 + ch];
            m = v > m ? v : m;
        }
        out_feat[(size_t)q * 64 + ch] = m;
    }
}

// ---------------------------------------------------------------------------
// Host-side launcher
// ---------------------------------------------------------------------------
extern "C" void kernel_launch(void* const* d_in, const int* in_sizes, int n_in,
                              void* d_out, int out_size, void* d_ws, size_t ws_size,
                              hipStream_t stream) {
    (void)in_sizes; (void)n_in; (void)out_size; (void)ws_size;

    const float* xyz  = (const float*)d_in[0];
    const float* feat = (const float*)d_in[1];
    const int*   sidx = (const int*)d_in[2];
    const float* w0  = (const float*)d_in[3];
    const float* b0  = (const float*)d_in[4];
    const float* g0  = (const float*)d_in[5];
    const float* bt0 = (const float*)d_in[6];
    const float* rm0 = (const float*)d_in[7];
    const float* rv0 = (const float*)d_in[8];
    const float* w1  = (const float*)d_in[9];
    const float* b1  = (const float*)d_in[10];
    const float* g1  = (const float*)d_in[11];
    const float* bt1 = (const float*)d_in[12];
    const float* rm1 = (const float*)d_in[13];
    const float* rv1 = (const float*)d_in[14];
    const float* w2  = (const float*)d_in[15];
    const float* b2  = (const float*)d_in[16];
    const float* g2  = (const float*)d_in[17];
    const float* bt2 = (const float*)d_in[18];
    const float* rm2 = (const float*)d_in[19];
    const float* rv2 = (const float*)d_in[20];

    // Output layout: new_xyz (B*S*3) | new_feature (B*S*64) | sample_idx (B*S)
    float* out      = (float*)d_out;
    float* out_xyz  = out;
    float* out_feat = out + (size_t)BB * SS * 3;
    float* out_idx  = out + (size_t)BB * SS * 3 + (size_t)BB * SS * 64;

    // Workspace layout
    char*  ws    = (char*)d_ws;
    float* pnorm = (float*)ws;  ws += (size_t)BB * NN * sizeof(float);
    float* qxyz  = (float*)ws;  ws += (size_t)BB * SS * 3 * sizeof(float);
    float* qnorm = (float*)ws;  ws += (size_t)BB * SS * sizeof(float);
    int*   topk  = (int*)ws;    // BB*SS*KK ints

    norms_kernel<<<(BB * NN + 255) / 256, 256, 0, stream>>>(xyz, pnorm);
    gather_kernel<<<(BB * SS + 255) / 256, 256, 0, stream>>>(xyz, sidx, qxyz, qnorm,
                                                             out_xyz, out_idx);
    topk_kernel<<<BB * SS, 256, 0, stream>>>(xyz, pnorm, qxyz, qnorm, topk);
    mlp_kernel<<<BB * SS, 32, 0, stream>>>(xyz, feat, topk, qxyz,
                                           w0, b0, g0, bt0, rm0, rv0,
                                           w1, b1, g1, bt1, rm1, rv1,
                                           w2, b2, g2, bt2, rm2, rv2,
                                           out_feat);
}